// GCLayer_20186346291917
// MI455X (gfx1250) — compile-verified
//
#include <hip/hip_runtime.h>
#include <hip/hip_bf16.h>
#include <math.h>

#define NN 1024
#define DD 64

typedef __attribute__((ext_vector_type(2))) float v2f;
typedef __attribute__((ext_vector_type(8))) float v8f;

#if defined(__HIP_DEVICE_COMPILE__) && __has_builtin(__builtin_amdgcn_wmma_f32_16x16x4_f32)
#define USE_WMMA_F32 1
#else
#define USE_WMMA_F32 0
#endif

// ---------------------------------------------------------------------------
// Generic 1024x64 @ 64x64 GEMM: C = act(A @ Bw + bias)
// One wave (32 threads) per 16x16 output tile; K-loop of 16 f32 WMMAs.
// A: M x 64 row-major. Bw: 64 x 64 row-major (ldb=64). bias: per-column or null.
// ---------------------------------------------------------------------------
__global__ void gemm1024x64x64(const float* __restrict__ A,
                               const float* __restrict__ Bw,
                               const float* __restrict__ bias,
                               float* __restrict__ C,
                               int relu)
{
    const int lane  = threadIdx.x;      // 0..31 (wave32)
    const int tileM = blockIdx.x * 16;
    const int tileN = blockIdx.y * 16;
    const int lm    = lane & 15;
    const int lh    = lane >> 4;        // 0 or 1

    float accs[8];

#if USE_WMMA_F32
    v8f acc = {};
    const float* Arow = A + (size_t)(tileM + lm) * DD;
    #pragma unroll
    for (int k0 = 0; k0 < DD; k0 += 4) {
        const int ka = k0 + lh * 2;
        v2f a, b;
        // A 16x4 tile: lanes 0-15 hold K=k0..k0+1, lanes 16-31 hold K=k0+2..k0+3
        a.x = Arow[ka];
        a.y = Arow[ka + 1];
        // B 4x16 tile (mirror of A over N): lane n holds B[ka][n], B[ka+1][n]
        b.x = Bw[(size_t)ka       * DD + tileN + lm];
        b.y = Bw[(size_t)(ka + 1) * DD + tileN + lm];
        // 8 args: (neg_a, A, neg_b, B, c_mod, C, reuse_a, reuse_b)
        acc = __builtin_amdgcn_wmma_f32_16x16x4_f32(false, a, false, b,
                                                    (short)0, acc, false, false);
    }
    #pragma unroll
    for (int r = 0; r < 8; ++r) accs[r] = acc[r];
#else
    #pragma unroll
    for (int r = 0; r < 8; ++r) accs[r] = 0.f;
    const int colf = tileN + lm;
    for (int k = 0; k < DD; ++k) {
        const float bv = Bw[(size_t)k * DD + colf];
        #pragma unroll
        for (int r = 0; r < 8; ++r) {
            const int row = tileM + r + 8 * lh;
            accs[r] = fmaf(A[(size_t)row * DD + k], bv, accs[r]);
        }
    }
#endif

    // C/D layout: VGPR r -> M = r + 8*lh, N = lane&15
    #pragma unroll
    for (int r = 0; r < 8; ++r) {
        const int row = tileM + r + 8 * lh;
        const int col = tileN + lm;
        float v = accs[r];
        if (bias) v += bias[col];
        if (relu) v = fmaxf(v, 0.f);
        C[(size_t)row * DD + col] = v;
    }
}

// ---------------------------------------------------------------------------
// Attention pass (the only O(N^2) work after algebraic refactoring):
//   att[i,j] = sigmoid( dot(relu(QL[i]+QR[j]+adj*wa), Wa2) + ba2 )
//   S[i]     = sum_j att[i,j] * relu(P[j] + adj[i,j]*wm)
//   ssum[i]  = sum_j att[i,j]
// One block per row i; 8 waves split j; each lane owns 2 of the 64 d-values.
// ---------------------------------------------------------------------------
__global__ __launch_bounds__(256) void attn_kernel(
    const float* __restrict__ adj, const float* __restrict__ P,
    const float* __restrict__ QL,  const float* __restrict__ QR,
    const float* __restrict__ wm,  const float* __restrict__ wa,
    const float* __restrict__ Wa2, const float* __restrict__ ba2,
    float* __restrict__ S, float* __restrict__ ssum)
{
    const int i    = blockIdx.x;
    const int lane = threadIdx.x & 31;
    const int wave = threadIdx.x >> 5;   // 0..7
    const int d0   = lane * 2;

    __shared__ float sS[DD];
    __shared__ float sAtt;
    if (threadIdx.x < DD) sS[threadIdx.x] = 0.f;
    if (threadIdx.x == 0) sAtt = 0.f;
    __syncthreads();

    const float ql0 = QL[(size_t)i * DD + d0];
    const float ql1 = QL[(size_t)i * DD + d0 + 1];
    const float wa0 = wa[d0],  wa1 = wa[d0 + 1];
    const float wm0 = wm[d0],  wm1 = wm[d0 + 1];
    const float w20 = Wa2[d0], w21 = Wa2[d0 + 1];
    const float b2  = ba2[0];

    float s0 = 0.f, s1 = 0.f, asum = 0.f;
    const float* adjrow = adj + (size_t)i * NN;

    const int j0 = wave * 128;
    for (int jc = j0; jc < j0 + 128; jc += 32) {
        const float av = adjrow[jc + lane];
        #pragma unroll 4
        for (int t = 0; t < 32; ++t) {
            const float a = __shfl(av, t, 32);
            const int   j = jc + t;
            const float2 qr = *(const float2*)(QR + (size_t)j * DD + d0);
            const float t10 = fmaxf(0.f, ql0 + qr.x + a * wa0);
            const float t11 = fmaxf(0.f, ql1 + qr.y + a * wa1);
            float part = t10 * w20 + t11 * w21;
            #pragma unroll
            for (int off = 16; off > 0; off >>= 1)
                part += __shfl_xor(part, off, 32);
            const float att = 1.f / (1.f + __expf(-(part + b2)));
            const float2 pj = *(const float2*)(P + (size_t)j * DD + d0);
            const float t20 = fmaxf(0.f, pj.x + a * wm0);
            const float t21 = fmaxf(0.f, pj.y + a * wm1);
            s0 = fmaf(att, t20, s0);
            s1 = fmaf(att, t21, s1);
            asum += att;
        }
    }

    atomicAdd(&sS[d0], s0);
    atomicAdd(&sS[d0 + 1], s1);
    if (lane == 0) atomicAdd(&sAtt, asum);
    __syncthreads();

    if (threadIdx.x < DD) S[(size_t)i * DD + threadIdx.x] = sS[threadIdx.x];
    if (threadIdx.x == 0) ssum[i] = sAtt;
}

// msgp[i][d] += ssum[i] * bm2[d]   (folds the att-weighted bm2 bias)
__global__ void add_scaled_bias(float* __restrict__ msgp,
                                const float* __restrict__ ssum,
                                const float* __restrict__ bm2)
{
    const int idx = blockIdx.x * 256 + threadIdx.x;   // NN*DD threads
    const int i = idx >> 6, d = idx & 63;
    msgp[idx] += ssum[i] * bm2[d];
}

// out = relu(layernorm(h + msg) * g + beta). One wave per row, 2 d per lane.
__global__ __launch_bounds__(256) void add_ln_relu(
    const float* __restrict__ h, const float* __restrict__ msg,
    const float* __restrict__ g, const float* __restrict__ beta,
    float* __restrict__ out)
{
    const int wave = threadIdx.x >> 5;
    const int lane = threadIdx.x & 31;
    const int i    = blockIdx.x * 8 + wave;
    const int d0   = lane * 2;

    float v0 = h[(size_t)i * DD + d0]     + msg[(size_t)i * DD + d0];
    float v1 = h[(size_t)i * DD + d0 + 1] + msg[(size_t)i * DD + d0 + 1];

    float s = v0 + v1;
    #pragma unroll
    for (int off = 16; off > 0; off >>= 1) s += __shfl_xor(s, off, 32);
    const float mu = s * (1.f / 64.f);

    const float e0 = v0 - mu, e1 = v1 - mu;
    float q = e0 * e0 + e1 * e1;
    #pragma unroll
    for (int off = 16; off > 0; off >>= 1) q += __shfl_xor(q, off, 32);
    const float rstd = rsqrtf(q * (1.f / 64.f) + 1e-5f);

    const float o0 = e0 * rstd * g[d0]     + beta[d0];
    const float o1 = e1 * rstd * g[d0 + 1] + beta[d0 + 1];
    out[(size_t)i * DD + d0]     = fmaxf(0.f, o0);
    out[(size_t)i * DD + d0 + 1] = fmaxf(0.f, o1);
}

extern "C" void kernel_launch(void* const* d_in, const int* in_sizes, int n_in,
                              void* d_out, int out_size, void* d_ws, size_t ws_size,
                              hipStream_t stream)
{
    const float* x    = (const float*)d_in[0];
    const float* adj  = (const float*)d_in[1];
    const float* Wl   = (const float*)d_in[2];
    const float* bl   = (const float*)d_in[3];
    const float* Wm1  = (const float*)d_in[4];   // 65 x 64
    const float* bm1  = (const float*)d_in[5];
    const float* Wm2  = (const float*)d_in[6];
    const float* bm2  = (const float*)d_in[7];
    const float* Wa1  = (const float*)d_in[8];   // 129 x 64
    const float* ba1  = (const float*)d_in[9];
    const float* Wa2  = (const float*)d_in[10];  // 64 x 1
    const float* ba2  = (const float*)d_in[11];
    const float* Wo1  = (const float*)d_in[12];
    const float* bo1  = (const float*)d_in[13];
    const float* Wo2  = (const float*)d_in[14];
    const float* bo2  = (const float*)d_in[15];
    const float* g    = (const float*)d_in[16];
    const float* beta = (const float*)d_in[17];

    float* ws   = (float*)d_ws;
    float* h    = ws;               // N*D
    float* P    = ws + 65536;       // N*D
    float* QL   = ws + 131072;      // N*D
    float* QR   = ws + 196608;      // N*D
    float* S    = ws + 262144;      // N*D
    float* ssum = ws + 327680;      // N
    float* msgp = QL;               // reuse after attention
    float* tb   = QR;               // reuse after attention
    float* msg2 = P;                // reuse after attention

    const dim3 ggrid(NN / 16, DD / 16), gblk(32);

    // h = x @ Wl + bl
    gemm1024x64x64<<<ggrid, gblk, 0, stream>>>(x, Wl, bl, h, 0);
    // P  = h @ Wm1[:64] + bm1 ; QL = h @ Wa1[:64] ; QR = h @ Wa1[64:128] + ba1
    gemm1024x64x64<<<ggrid, gblk, 0, stream>>>(h, Wm1, bm1, P, 0);
    gemm1024x64x64<<<ggrid, gblk, 0, stream>>>(h, Wa1, nullptr, QL, 0);
    gemm1024x64x64<<<ggrid, gblk, 0, stream>>>(h, Wa1 + 64 * DD, ba1, QR, 0);

    // O(N^2) attention-weighted reduction
    attn_kernel<<<NN, 256, 0, stream>>>(adj, P, QL, QR,
                                        Wm1 + 64 * DD, Wa1 + 128 * DD,
                                        Wa2, ba2, S, ssum);

    // msg = relu((S@Wm2 + ssum*bm2) @ Wo1 + bo1) @ Wo2 + bo2
    gemm1024x64x64<<<ggrid, gblk, 0, stream>>>(S, Wm2, nullptr, msgp, 0);
    add_scaled_bias<<<NN * DD / 256, 256, 0, stream>>>(msgp, ssum, bm2);
    gemm1024x64x64<<<ggrid, gblk, 0, stream>>>(msgp, Wo1, bo1, tb, 1);
    gemm1024x64x64<<<ggrid, gblk, 0, stream>>>(tb, Wo2, bo2, msg2, 0);

    // out = relu(layernorm(h + msg)); second output = adj passthrough
    add_ln_relu<<<NN / 8, 256, 0, stream>>>(h, msg2, g, beta, (float*)d_out);
    hipMemcpyAsync((float*)d_out + NN * DD, adj, (size_t)NN * NN * sizeof(float),
                   hipMemcpyDeviceToDevice, stream);
}